// MSE_SSIM_Loss_36232344109639
// MI455X (gfx1250) — compile-verified
//
#include <hip/hip_runtime.h>
#include <hip/hip_bf16.h>

typedef float v2f __attribute__((ext_vector_type(2)));
typedef float v8f __attribute__((ext_vector_type(8)));

// Normalized 11-tap Gaussian (sigma=1.5), precomputed at full precision.
__device__ const float GW[11] = {
    0.001028378f, 0.007598700f, 0.036000800f, 0.109360900f, 0.213005700f,
    0.266011800f,
    0.213005700f, 0.109360900f, 0.036000800f, 0.007598700f, 0.001028378f
};

#define IMG_H   512
#define IMG_W   512
#define OUT_HW  502
#define BATCH   32
#define N_ELEMS (BATCH * IMG_H * IMG_W)       // 8388608
#define N_SSIM  (BATCH * OUT_HW * OUT_HW)     // 8064128

#define MSE_BLOCKS 1024
#define SSIM_BLOCKS (8 * 16 * 32)             // 4096

// ---- per-WG SSIM tiling ----
// Block covers 32x64 outputs = 8 waves x (16x16 WMMA tiles in 2x4 layout).
// Horizontal-blur staging: 44 rows x 64 cols x 5 channels in LDS.
#define HB_ROWS 44
#define HB_COLS 64
#define HB_LD   65                            // pad to dodge bank conflicts
#define HB_CH   (HB_ROWS * HB_LD)             // 2860 floats / channel

// Zero-padded Gaussian band table: s_band[d + BAND_OFF] = GW[d] for d in [0,10].
#define BAND_OFF  16
#define BAND_LEN  48

// ---------------- Kernel 1: MSE partial sums ----------------
__global__ void mse_partial_kernel(const float* __restrict__ p,
                                   const float* __restrict__ t,
                                   float* __restrict__ partials) {
    __shared__ float red[256];
    const int tid = threadIdx.x;
    float s = 0.0f;
    for (size_t i = (size_t)blockIdx.x * 256 + tid; i < (size_t)N_ELEMS;
         i += (size_t)MSE_BLOCKS * 256) {
        float d = p[i] - t[i];
        s += d * d;
    }
    red[tid] = s;
    __syncthreads();
    for (int off = 128; off > 0; off >>= 1) {
        if (tid < off) red[tid] += red[tid + off];
        __syncthreads();
    }
    if (tid == 0) partials[blockIdx.x] = red[0];
}

// ---------------- Kernel 2: SSIM via WMMA vertical blur ----------------
// grid = (8 col-groups, 16 row-groups, 32 images), block = 256 threads.
__global__ void ssim_wmma_kernel(const float* __restrict__ pred,
                                 const float* __restrict__ target,
                                 float* __restrict__ partials) {
    __shared__ float s_hb[5 * HB_CH];   // 57,200 B
    __shared__ float s_band[BAND_LEN];  // padded Gaussian band for A frags
    __shared__ float s_red[8];

    const int tid = threadIdx.x;
    const int c0 = blockIdx.x * 64;     // output col origin
    const int r0 = blockIdx.y * 32;     // output row origin
    const int b  = blockIdx.z;

    const float* pimg = pred   + (size_t)b * IMG_H * IMG_W;
    const float* timg = target + (size_t)b * IMG_H * IMG_W;

    // ---- Phase 0: fill the padded band table (no divergence later) ----
    if (tid < BAND_LEN) {
        const int d = tid - BAND_OFF;
        s_band[tid] = (d >= 0 && d <= 10) ? GW[d] : 0.0f;
    }

    // ---- Phase 1: horizontal 11-tap blur of the 5 channels into LDS ----
    for (int idx = tid; idx < HB_ROWS * HB_COLS; idx += 256) {
        const int r  = idx >> 6;        // 0..43
        const int c  = idx & 63;        // 0..63
        const int gr = min(r0 + r, IMG_H - 1);
        const float* prow = pimg + (size_t)gr * IMG_W;
        const float* trow = timg + (size_t)gr * IMG_W;
        float s1 = 0.f, s2 = 0.f, s3 = 0.f, s4 = 0.f, s5 = 0.f;
#pragma unroll
        for (int k = 0; k < 11; k++) {
            const int gc = min(c0 + c + k, IMG_W - 1);
            const float pv = prow[gc];
            const float tv = trow[gc];
            const float wk = GW[k];     // constant k -> folded immediates
            s1 += wk * pv;
            s2 += wk * tv;
            s3 += wk * pv * pv;
            s4 += wk * tv * tv;
            s5 += wk * pv * tv;
        }
        const int o = r * HB_LD + c;
        s_hb[0 * HB_CH + o] = s1;
        s_hb[1 * HB_CH + o] = s2;
        s_hb[2 * HB_CH + o] = s3;
        s_hb[3 * HB_CH + o] = s4;
        s_hb[4 * HB_CH + o] = s5;
    }
    __syncthreads();

    // ---- Phase 2: vertical blur via V_WMMA_F32_16X16X4_F32 ----
    const int wv   = tid >> 5;          // wave id 0..7
    const int lane = tid & 31;
    const int TR   = (wv >> 2) * 16;    // tile row offset within block (0/16)
    const int TC   = (wv & 3)  * 16;    // tile col offset within block
    const int half = lane >> 4;         // 0: lanes 0-15, 1: lanes 16-31
    const int ln   = lane & 15;

    // Pre-load all 7 A fragments from the LDS band table (plain ds_loads,
    // EXEC stays all-ones through the entire WMMA region).
    v2f afrag[7];
#pragma unroll
    for (int i = 0; i < 7; i++) {
        const int kb = 4 * i + 2 * half;
        afrag[i].x = s_band[kb     - ln + BAND_OFF];  // A[m=ln][k=kb]
        afrag[i].y = s_band[kb + 1 - ln + BAND_OFF];  // A[m=ln][k=kb+1]
    }

    v8f acc[5];
#pragma unroll
    for (int ch = 0; ch < 5; ch++) acc[ch] = (v8f){0,0,0,0,0,0,0,0};

    // D = A(16x28 banded Gaussian) x B(28x16 hblur cols), 7 chunks of K=4.
#pragma unroll
    for (int i = 0; i < 7; i++) {
        const int kb = 4 * i + 2 * half;
#pragma unroll
        for (int ch = 0; ch < 5; ch++) {
            const float* hbc = &s_hb[ch * HB_CH];
            v2f bfrag;
            bfrag.x = hbc[(TR + kb    ) * HB_LD + TC + ln];  // B[k=kb][n=ln]
            bfrag.y = hbc[(TR + kb + 1) * HB_LD + TC + ln];
            acc[ch] = __builtin_amdgcn_wmma_f32_16x16x4_f32(
                false, afrag[i], false, bfrag, (short)0, acc[ch], false, false);
        }
    }

    // ---- Phase 3: SSIM map + deterministic reduction ----
    const float C1 = 1e-4f;   // (0.01)^2
    const float C2 = 9e-4f;   // (0.03)^2
    float local = 0.0f;
#pragma unroll
    for (int v = 0; v < 8; v++) {
        const int m    = v + 8 * half;      // D row (C/D VGPR layout)
        const int orow = r0 + TR + m;
        const int ocol = c0 + TC + ln;
        if (orow < OUT_HW && ocol < OUT_HW) {
            const float mu1   = acc[0][v];
            const float mu2   = acc[1][v];
            const float mu1sq = mu1 * mu1;
            const float mu2sq = mu2 * mu2;
            const float mu12  = mu1 * mu2;
            const float s1    = acc[2][v] - mu1sq;
            const float s2    = acc[3][v] - mu2sq;
            const float s12   = acc[4][v] - mu12;
            const float num   = (2.0f * mu12 + C1) * (2.0f * s12 + C2);
            const float den   = (mu1sq + mu2sq + C1) * (s1 + s2 + C2) + 1e-6f;
            local += num / den;
        }
    }
    // wave32 shuffle reduction
    for (int off = 16; off > 0; off >>= 1)
        local += __shfl_down(local, off, 32);
    if (lane == 0) s_red[wv] = local;
    __syncthreads();
    if (tid == 0) {
        float s = 0.0f;
#pragma unroll
        for (int w = 0; w < 8; w++) s += s_red[w];
        const int pid = (blockIdx.z * 16 + blockIdx.y) * 8 + blockIdx.x;
        partials[pid] = s;
    }
}

// ---------------- Kernel 3: final deterministic combine ----------------
__global__ void finalize_kernel(const float* __restrict__ ws,
                                float* __restrict__ out) {
    __shared__ float red[256];
    __shared__ float mse_sum_s;
    const int tid = threadIdx.x;

    float s = 0.0f;
    for (int i = tid; i < MSE_BLOCKS; i += 256) s += ws[i];
    red[tid] = s;
    __syncthreads();
    for (int off = 128; off > 0; off >>= 1) {
        if (tid < off) red[tid] += red[tid + off];
        __syncthreads();
    }
    if (tid == 0) mse_sum_s = red[0];
    __syncthreads();

    s = 0.0f;
    for (int i = tid; i < SSIM_BLOCKS; i += 256) s += ws[MSE_BLOCKS + i];
    red[tid] = s;
    __syncthreads();
    for (int off = 128; off > 0; off >>= 1) {
        if (tid < off) red[tid] += red[tid + off];
        __syncthreads();
    }
    if (tid == 0) {
        const float mse  = mse_sum_s / (float)N_ELEMS;
        const float ssim = red[0] / (float)N_SSIM;
        out[0] = 0.6f * mse + 0.4f * (1.0f - ssim);
    }
}

extern "C" void kernel_launch(void* const* d_in, const int* in_sizes, int n_in,
                              void* d_out, int out_size, void* d_ws, size_t ws_size,
                              hipStream_t stream) {
    const float* pred   = (const float*)d_in[0];
    const float* target = (const float*)d_in[1];
    float* out = (float*)d_out;
    float* ws  = (float*)d_ws;   // [0,1024): mse partials; [1024,5120): ssim partials

    mse_partial_kernel<<<MSE_BLOCKS, 256, 0, stream>>>(pred, target, ws);
    ssim_wmma_kernel<<<dim3(8, 16, 32), 256, 0, stream>>>(pred, target,
                                                          ws + MSE_BLOCKS);
    finalize_kernel<<<1, 256, 0, stream>>>(ws, out);
}